// MDHR_25864293056859
// MI455X (gfx1250) — compile-verified
//
#include <hip/hip_runtime.h>

#define C      512
#define NC     12
#define BT_TOT 1024
#define SEQ    21
#define NROWS  (BT_TOT * SEQ)   // 21504
#define EPSV   1e-5f

#define KSTRIDE 34              // LDS k-stride per column (even -> 8B-aligned pairs; bank-safe)
#define BSLICE  (64 * KSTRIDE)  // one 32-K x 64-N B slice in LDS, column-major (floats)

typedef __attribute__((ext_vector_type(2))) float v2f;
typedef __attribute__((ext_vector_type(8))) float v8f;

typedef __attribute__((address_space(1))) int* as1_ip;
typedef __attribute__((address_space(3))) int* as3_ip;

#if defined(__has_builtin)
#  if __has_builtin(__builtin_amdgcn_global_load_async_to_lds_b32)
#    define HAS_ASYNC_LDS 1
#  else
#    define HAS_ASYNC_LDS 0
#  endif
#else
#  define HAS_ASYNC_LDS 0
#endif

#if HAS_ASYNC_LDS
#  if __has_builtin(__builtin_amdgcn_s_wait_asynccnt)
#    define WAIT_ASYNC(n) __builtin_amdgcn_s_wait_asynccnt(n)
#  else
#    define WAIT_ASYNC(n) asm volatile("s_wait_asynccnt " #n ::: "memory")
#  endif
#endif

__constant__ int   d_SEL[NC]  = {0,0,0,1,0,3,2,2,2,3,3,3};
__constant__ int   d_ROFF[4]  = {0,14,28,28};
__constant__ float d_ADJ[NC*NC] = {
    0,0,0,1,0,1,1,1,1,1,1,1,
    0,0,0,1,0,1,1,1,1,1,1,1,
    0,0,0,1,0,1,1,1,1,1,1,1,
    1,1,1,0,1,1,1,1,1,1,1,1,
    0,0,0,1,0,1,1,1,1,1,1,1,
    1,1,1,1,1,0,1,1,1,0,0,0,
    1,1,1,1,1,1,0,0,0,1,1,1,
    1,1,1,1,1,1,0,0,0,1,1,1,
    1,1,1,1,1,1,0,0,0,1,1,1,
    1,1,1,1,1,0,1,1,1,0,0,0,
    1,1,1,1,1,0,1,1,1,0,0,0,
    1,1,1,1,1,0,1,1,1,0,0,0};

// ---------------------------------------------------------------------------
// GEMM core: block = 256 threads (8 waves), tile M=128 x N=64, K=512.
// Each wave owns 16 rows x 64 cols (4 x v8f accumulators, A-frag reuse x4).
// B slices (32K x 64N) staged COLUMN-MAJOR in LDS so each WMMA B fragment
// (k,k+1 of one column) is one aligned ds_load_b64 into an adjacent VGPR pair.
// Stage uses GLOBAL_LOAD_ASYNC_TO_LDS_B32 double-buffered on ASYNCcnt.
// WMMA layouts per CDNA5 ISA §7.12.2 (A 16x4 / B 4x16 / C 16x16 f32).
// ---------------------------------------------------------------------------
__device__ inline void gemm_issue_sliceB(const float* __restrict__ W, int n0,
                                         int s, float* __restrict__ ldsB)
{
#if HAS_ASYNC_LDS
    const int tid = threadIdx.x;
    float* buf = ldsB + (s & 1) * BSLICE;
#pragma unroll
    for (int i = 0; i < 8; ++i) {
        int e   = tid + i * 256;        // 0..2047
        int col = e & 63;
        int k   = e >> 6;               // 0..31
        const float* src = W + (size_t)(s * 32 + k) * C + n0 + col;
        float* dst = buf + col * KSTRIDE + k;
        __builtin_amdgcn_global_load_async_to_lds_b32((as1_ip)src, (as3_ip)dst, 0, 0);
    }
#else
    (void)W; (void)n0; (void)s; (void)ldsB;
#endif
}

__device__ inline void gemm_slice_compute(const float* __restrict__ arow, int kb,
                                          const float* __restrict__ bB,
                                          int ksel, int nlo, v8f acc[4])
{
#pragma unroll
    for (int ks = 0; ks < 32; ks += 4) {
        v2f a;
        a.x = arow[kb + ks + ksel];
        a.y = arow[kb + ks + ksel + 1];
        const float* bp = bB + (size_t)nlo * KSTRIDE + ks + ksel;
        v2f b0 = *(const v2f*)(bp);
        v2f b1 = *(const v2f*)(bp + 16 * KSTRIDE);
        v2f b2 = *(const v2f*)(bp + 32 * KSTRIDE);
        v2f b3 = *(const v2f*)(bp + 48 * KSTRIDE);
        acc[0] = __builtin_amdgcn_wmma_f32_16x16x4_f32(false, a, false, b0, (short)0, acc[0], false, false);
        acc[1] = __builtin_amdgcn_wmma_f32_16x16x4_f32(false, a, false, b1, (short)0, acc[1], false, false);
        acc[2] = __builtin_amdgcn_wmma_f32_16x16x4_f32(false, a, false, b2, (short)0, acc[2], false, false);
        acc[3] = __builtin_amdgcn_wmma_f32_16x16x4_f32(false, a, false, b3, (short)0, acc[3], false, false);
    }
}

__device__ inline void gemm_core128(const float* __restrict__ arow,
                                    const float* __restrict__ W, int n0,
                                    float* __restrict__ ldsB, v8f acc[4])
{
    const int lane = threadIdx.x & 31;
    const int ksel = (lane >> 4) << 1;   // 0 or 2
    const int nlo  = lane & 15;

#if HAS_ASYNC_LDS
    gemm_issue_sliceB(W, n0, 0, ldsB);
    gemm_issue_sliceB(W, n0, 1, ldsB);
#pragma unroll 1
    for (int s = 0; s < 14; ++s) {
        WAIT_ASYNC(8);
        __syncthreads();
        gemm_slice_compute(arow, s * 32, ldsB + (s & 1) * BSLICE, ksel, nlo, acc);
        __syncthreads();
        gemm_issue_sliceB(W, n0, s + 2, ldsB);
    }
    WAIT_ASYNC(8);
    __syncthreads();
    gemm_slice_compute(arow, 14 * 32, ldsB, ksel, nlo, acc);     // slice 14 -> buf 0
    __syncthreads();
    WAIT_ASYNC(0);
    __syncthreads();
    gemm_slice_compute(arow, 15 * 32, ldsB + BSLICE, ksel, nlo, acc);
#else
    const int tid = threadIdx.x;
#pragma unroll 1
    for (int s = 0; s < 16; ++s) {
        __syncthreads();
#pragma unroll
        for (int i = 0; i < 8; ++i) {
            int e   = tid + i * 256;
            int col = e & 63;
            int k   = e >> 6;
            ldsB[col * KSTRIDE + k] = W[(size_t)(s * 32 + k) * C + n0 + col];
        }
        __syncthreads();
        gemm_slice_compute(arow, s * 32, ldsB, ksel, nlo, acc);
    }
#endif
}

// ---------------------------------------------------------------------------
// 1) Region GEMM: hraw[r] = regions[r] @ region_W[r] + b ; accumulate BN stats
// grid (NROWS/128, C/64, 4), block 256
// ---------------------------------------------------------------------------
__global__ void k_region_gemm(const float* __restrict__ x,
                              const float* __restrict__ rW,
                              const float* __restrict__ rb,
                              float* __restrict__ hbuf,
                              float* __restrict__ rstats)
{
    __shared__ float ldsB[2 * BSLICE];
    const int r    = blockIdx.z;
    const int wave = threadIdx.x >> 5;
    const int lane = threadIdx.x & 31;
    const int n0   = blockIdx.y * 64;

    const int myrow = blockIdx.x * 128 + wave * 16 + (lane & 15);
    const int bt    = myrow / SEQ;
    const int s     = myrow - bt * SEQ;
    const float* arow = x + ((size_t)bt * 49 + d_ROFF[r] + s) * C;
    const float* W    = rW + (size_t)r * C * C;

    v8f acc[4] = {};
    gemm_core128(arow, W, n0, ldsB, acc);

    const int nlo  = lane & 15;
    const int rowb = blockIdx.x * 128 + wave * 16 + ((lane >> 4) << 3);
#pragma unroll
    for (int nt = 0; nt < 4; ++nt) {
        const int cc = n0 + nt * 16 + nlo;
        const float bias = rb[r * C + cc];
        float s0 = 0.f, q0 = 0.f;
#pragma unroll
        for (int v = 0; v < 8; ++v) {
            float y = acc[nt][v] + bias;
            hbuf[((size_t)r * NROWS + (size_t)(rowb + v)) * C + cc] = y;
            s0 += y; q0 += y * y;
        }
        atomicAdd(&rstats[(r * C + cc) * 2],     s0);
        atomicAdd(&rstats[(r * C + cc) * 2 + 1], q0);
    }
}

// ---------------------------------------------------------------------------
// 2) Region BN + ReLU (in place) + fused mean over seq -> hmean[4,BT,C]
// ---------------------------------------------------------------------------
__global__ void k_region_finalize(float* __restrict__ hbuf,
                                  const float* __restrict__ rstats,
                                  const float* __restrict__ gamma,
                                  const float* __restrict__ beta,
                                  float* __restrict__ hmean)
{
    int idx = blockIdx.x * blockDim.x + threadIdx.x;
    int c   = idx & (C - 1);
    int bt  = (idx >> 9) & 1023;
    int r   = idx >> 19;
    float sum  = rstats[(r * C + c) * 2];
    float sq   = rstats[(r * C + c) * 2 + 1];
    float mean = sum / (float)NROWS;
    float var  = sq / (float)NROWS - mean * mean;
    float rstd = rsqrtf(var + EPSV);
    float ga = gamma[r * C + c], be = beta[r * C + c];
    float acc = 0.f;
    size_t base = ((size_t)r * NROWS + (size_t)bt * SEQ) * C + c;
    for (int s2 = 0; s2 < SEQ; ++s2) {
        float vv = hbuf[base + (size_t)s2 * C];
        float hn = ga * (vv - mean) * rstd + be;
        hn = hn > 0.f ? hn : 0.f;
        hbuf[base + (size_t)s2 * C] = hn;
        acc += hn;
    }
    hmean[((size_t)r * BT_TOT + bt) * C + c] = acc * (1.f / SEQ);
}

// ---------------------------------------------------------------------------
// 3) Four small FC heads from hmean -> first 43008 floats of d_out
// ---------------------------------------------------------------------------
__global__ void k_heads(const float* __restrict__ hmean,
                        const float* __restrict__ upW,  const float* __restrict__ upb,
                        const float* __restrict__ midW, const float* __restrict__ midb,
                        const float* __restrict__ d1W,  const float* __restrict__ d1b,
                        const float* __restrict__ d2W,  const float* __restrict__ d2b,
                        float* __restrict__ out)
{
    int bt = blockIdx.x;
    int j  = threadIdx.x;
    int r, col, ncol; const float* W; const float* b; size_t ooff;
    if (j < 16)      { r = 0; col = j;      ncol = 16; W = upW;  b = upb;  ooff = 0     + (size_t)bt * 16; }
    else if (j < 18) { r = 1; col = j - 16; ncol = 2;  W = midW; b = midb; ooff = 16384 + (size_t)bt * 2;  }
    else if (j < 26) { r = 2; col = j - 18; ncol = 8;  W = d1W;  b = d1b;  ooff = 18432 + (size_t)bt * 8;  }
    else if (j < 42) { r = 3; col = j - 26; ncol = 16; W = d2W;  b = d2b;  ooff = 26624 + (size_t)bt * 16; }
    else return;
    const float* hv = hmean + ((size_t)r * BT_TOT + bt) * C;
    float acc = b[col];
    for (int c = 0; c < C; ++c) acc += hv[c] * W[c * ncol + col];
    out[ooff + col] = acc;
}

// ---------------------------------------------------------------------------
// 4) Class GEMM pass 1: BN stats only (y recomputed in pass 2)
// grid (NROWS/128, C/64, 12), block 256
// ---------------------------------------------------------------------------
__global__ void k_class_pass1(const float* __restrict__ hbuf,
                              const float* __restrict__ cW,
                              const float* __restrict__ cb,
                              float* __restrict__ cstats)
{
    __shared__ float ldsB[2 * BSLICE];
    const int n    = blockIdx.z;
    const int wave = threadIdx.x >> 5;
    const int lane = threadIdx.x & 31;
    const int n0   = blockIdx.y * 64;

    const int myrow = blockIdx.x * 128 + wave * 16 + (lane & 15);
    const float* arow = hbuf + ((size_t)d_SEL[n] * NROWS + myrow) * C;
    const float* W    = cW + (size_t)n * C * C;

    v8f acc[4] = {};
    gemm_core128(arow, W, n0, ldsB, acc);

    const int nlo = lane & 15;
#pragma unroll
    for (int nt = 0; nt < 4; ++nt) {
        const int cc = n0 + nt * 16 + nlo;
        const float bias = cb[n * C + cc];
        float s0 = 0.f, q0 = 0.f;
#pragma unroll
        for (int v = 0; v < 8; ++v) {
            float y = acc[nt][v] + bias;
            s0 += y; q0 += y * y;
        }
        atomicAdd(&cstats[(n * C + cc) * 2],     s0);
        atomicAdd(&cstats[(n * C + cc) * 2 + 1], q0);
    }
}

// ---------------------------------------------------------------------------
// 5) Class GEMM pass 2: recompute, BN+ReLU, fused mean over seq -> f_v[BT,12,C]
// ---------------------------------------------------------------------------
__global__ void k_class_pass2(const float* __restrict__ hbuf,
                              const float* __restrict__ cW,
                              const float* __restrict__ cb,
                              const float* __restrict__ cstats,
                              const float* __restrict__ cgamma,
                              const float* __restrict__ cbeta,
                              float* __restrict__ f_v)
{
    __shared__ float ldsB[2 * BSLICE];
    const int n    = blockIdx.z;
    const int wave = threadIdx.x >> 5;
    const int lane = threadIdx.x & 31;
    const int n0   = blockIdx.y * 64;

    const int myrow = blockIdx.x * 128 + wave * 16 + (lane & 15);
    const float* arow = hbuf + ((size_t)d_SEL[n] * NROWS + myrow) * C;
    const float* W    = cW + (size_t)n * C * C;

    v8f acc[4] = {};
    gemm_core128(arow, W, n0, ldsB, acc);

    const int nlo  = lane & 15;
    const int rowb = blockIdx.x * 128 + wave * 16 + ((lane >> 4) << 3);
#pragma unroll
    for (int nt = 0; nt < 4; ++nt) {
        const int cc = n0 + nt * 16 + nlo;
        const float bias = cb[n * C + cc];
        float m0 = cstats[(n * C + cc) * 2] / (float)NROWS;
        float v0 = cstats[(n * C + cc) * 2 + 1] / (float)NROWS - m0 * m0;
        float r0 = rsqrtf(v0 + EPSV);
        float ga = cgamma[n * C + cc], be = cbeta[n * C + cc];
#pragma unroll
        for (int v = 0; v < 8; ++v) {
            int row = rowb + v;
            int bt  = row / SEQ;
            float y = ga * (acc[nt][v] + bias - m0) * r0 + be;
            y = y > 0.f ? y : 0.f;
            atomicAdd(&f_v[((size_t)bt * NC + n) * C + cc], y * (1.f / SEQ));
        }
    }
}

// ---------------------------------------------------------------------------
// 6) hg = f_v @ gat_W   (12288 x 512 @ 512 x 512), grid (96, 8), block 256
// ---------------------------------------------------------------------------
__global__ void k_hg_gemm(const float* __restrict__ f_v,
                          const float* __restrict__ gatW,
                          float* __restrict__ hg)
{
    __shared__ float ldsB[2 * BSLICE];
    const int wave = threadIdx.x >> 5;
    const int lane = threadIdx.x & 31;
    const int n0   = blockIdx.y * 64;

    const int myrow = blockIdx.x * 128 + wave * 16 + (lane & 15);
    const float* arow = f_v + (size_t)myrow * C;

    v8f acc[4] = {};
    gemm_core128(arow, gatW, n0, ldsB, acc);

    const int nlo  = lane & 15;
    const int rowb = blockIdx.x * 128 + wave * 16 + ((lane >> 4) << 3);
#pragma unroll
    for (int nt = 0; nt < 4; ++nt) {
        const int cc = n0 + nt * 16 + nlo;
#pragma unroll
        for (int v = 0; v < 8; ++v)
            hg[(size_t)(rowb + v) * C + cc] = acc[nt][v];
    }
}

// ---------------------------------------------------------------------------
// 7) 12-node GAT per sample: attention + residual. grid 1024, block 256
// ---------------------------------------------------------------------------
__global__ void k_gat(const float* __restrict__ f_v,
                      const float* __restrict__ hg,
                      const float* __restrict__ gal,
                      const float* __restrict__ gar,
                      const float* __restrict__ adj_mask,
                      float* __restrict__ g)
{
    __shared__ float s_al[NC], s_ar[NC];
    __shared__ float s_attn[NC][NC];
    int bt   = blockIdx.x;
    int wave = threadIdx.x >> 5;
    int lane = threadIdx.x & 31;
    const float* hgb = hg + (size_t)bt * NC * C;

    for (int d = wave; d < 2 * NC; d += 8) {
        int i = d >> 1;
        const float* vec = (d & 1) ? gar : gal;
        const float* hr  = hgb + (size_t)i * C;
        float p = 0.f;
        for (int c2 = lane; c2 < C; c2 += 32) p += hr[c2] * vec[c2];
        for (int off = 16; off > 0; off >>= 1) p += __shfl_down(p, off, 32);
        if (lane == 0) { if (d & 1) s_ar[i] = p; else s_al[i] = p; }
    }
    __syncthreads();

    if (threadIdx.x < NC * NC) {
        int i = threadIdx.x / NC, j = threadIdx.x % NC;
        float e = s_al[i] + s_ar[j];
        e = e > 0.f ? e : 0.2f * e;
        float adj = d_ADJ[i * NC + j] * adj_mask[(size_t)bt * NC * NC + i * NC + j]
                  + (i == j ? 1.f : 0.f);
        s_attn[i][j] = (adj > 0.1f) ? e : -1e9f;
    }
    __syncthreads();

    if (threadIdx.x < NC) {
        int i = threadIdx.x;
        float mx = -1e30f;
        for (int j = 0; j < NC; ++j) mx = fmaxf(mx, s_attn[i][j]);
        float ssum = 0.f;
        for (int j = 0; j < NC; ++j) { float ex = __expf(s_attn[i][j] - mx); s_attn[i][j] = ex; ssum += ex; }
        float inv = 1.f / ssum;
        for (int j = 0; j < NC; ++j) s_attn[i][j] *= inv;
    }
    __syncthreads();

    for (int c2 = threadIdx.x; c2 < C; c2 += blockDim.x) {
        float hv[NC];
#pragma unroll
        for (int j = 0; j < NC; ++j) hv[j] = hgb[(size_t)j * C + c2];
#pragma unroll
        for (int i = 0; i < NC; ++i) {
            float acc = f_v[((size_t)bt * NC + i) * C + c2];
#pragma unroll
            for (int j = 0; j < NC; ++j) acc += s_attn[i][j] * hv[j];
            g[((size_t)bt * NC + i) * C + c2] = acc;
        }
    }
}

// ---------------------------------------------------------------------------
// 8) Depthwise temporal conv (k=5, pad 2) + per-class stats
// ---------------------------------------------------------------------------
__global__ void k_tconv(const float* __restrict__ g,
                        const float* __restrict__ tW,
                        const float* __restrict__ tb,
                        float* __restrict__ vbuf,
                        float* __restrict__ tstats)
{
    int idx = blockIdx.x * 256 + threadIdx.x;
    int c  = idx & (C - 1);
    int n  = (idx >> 9) % NC;
    int bt = idx / (C * NC);
    int b  = bt >> 5;
    int t  = bt & 31;
    float acc = tb[n * C + c];
#pragma unroll
    for (int k = 0; k < 5; ++k) {
        int tt = t + k - 2;
        if (tt >= 0 && tt < 32)
            acc += g[(((size_t)(b * 32 + tt)) * NC + n) * C + c] * tW[(n * C + c) * 5 + k];
    }
    vbuf[idx] = acc;

    __shared__ float rsum[256], rsq[256];
    rsum[threadIdx.x] = acc;
    rsq[threadIdx.x]  = acc * acc;
    __syncthreads();
    for (int off = 128; off > 0; off >>= 1) {
        if (threadIdx.x < off) {
            rsum[threadIdx.x] += rsum[threadIdx.x + off];
            rsq[threadIdx.x]  += rsq[threadIdx.x + off];
        }
        __syncthreads();
    }
    if (threadIdx.x == 0) {
        atomicAdd(&tstats[n * 2],     rsum[0]);
        atomicAdd(&tstats[n * 2 + 1], rsq[0]);
    }
}

// ---------------------------------------------------------------------------
// 9) Final per-class BN + ReLU -> d_out tail
// ---------------------------------------------------------------------------
__global__ void k_final(const float* __restrict__ vbuf,
                        const float* __restrict__ tstats,
                        const float* __restrict__ tgamma,
                        const float* __restrict__ tbeta,
                        float* __restrict__ out)
{
    int idx = blockIdx.x * 256 + threadIdx.x;
    int n = (idx >> 9) % NC;
    const float N = (float)(BT_TOT * C);
    float mean = tstats[n * 2] / N;
    float var  = tstats[n * 2 + 1] / N - mean * mean;
    float rstd = rsqrtf(var + EPSV);
    float y = tgamma[n] * (vbuf[idx] - mean) * rstd + tbeta[n];
    out[idx] = y > 0.f ? y : 0.f;
}

// ---------------------------------------------------------------------------
extern "C" void kernel_launch(void* const* d_in, const int* in_sizes, int n_in,
                              void* d_out, int out_size, void* d_ws, size_t ws_size,
                              hipStream_t stream)
{
    (void)in_sizes; (void)n_in; (void)out_size; (void)ws_size;

    const float* x        = (const float*)d_in[0];
    const float* adj_mask = (const float*)d_in[1];
    const float* region_W = (const float*)d_in[2];
    const float* region_b = (const float*)d_in[3];
    const float* region_g = (const float*)d_in[4];
    const float* region_be= (const float*)d_in[5];
    const float* upfc_W   = (const float*)d_in[6];
    const float* upfc_b   = (const float*)d_in[7];
    const float* midfc_W  = (const float*)d_in[8];
    const float* midfc_b  = (const float*)d_in[9];
    const float* d1fc_W   = (const float*)d_in[10];
    const float* d1fc_b   = (const float*)d_in[11];
    const float* d2fc_W   = (const float*)d_in[12];
    const float* d2fc_b   = (const float*)d_in[13];
    const float* class_W  = (const float*)d_in[14];
    const float* class_b  = (const float*)d_in[15];
    const float* class_g  = (const float*)d_in[16];
    const float* class_be = (const float*)d_in[17];
    const float* gat_W    = (const float*)d_in[18];
    const float* gat_al   = (const float*)d_in[19];
    const float* gat_ar   = (const float*)d_in[20];
    const float* tconv_W  = (const float*)d_in[21];
    const float* tconv_b  = (const float*)d_in[22];
    const float* tbn_g    = (const float*)d_in[23];
    const float* tbn_be   = (const float*)d_in[24];

    float* out = (float*)d_out;
    float* ws  = (float*)d_ws;

    size_t o_hbuf   = 0;
    size_t o_rstats = o_hbuf   + (size_t)4 * NROWS * C;
    size_t o_cstats = o_rstats + (size_t)4 * C * 2;
    size_t o_hmean  = o_cstats + (size_t)NC * C * 2;
    size_t o_fv     = o_hmean  + (size_t)4 * BT_TOT * C;
    size_t o_hg     = o_fv     + (size_t)BT_TOT * NC * C;
    size_t o_g      = o_hg     + (size_t)BT_TOT * NC * C;
    size_t o_v      = o_g      + (size_t)BT_TOT * NC * C;
    size_t o_tstats = o_v      + (size_t)BT_TOT * NC * C;

    float* hbuf   = ws + o_hbuf;
    float* rstats = ws + o_rstats;
    float* cstats = ws + o_cstats;
    float* hmean  = ws + o_hmean;
    float* f_v    = ws + o_fv;
    float* hg     = ws + o_hg;
    float* gbuf   = ws + o_g;
    float* vbuf   = ws + o_v;
    float* tstats = ws + o_tstats;

    (void)hipMemsetAsync(rstats, 0, (o_hg - o_rstats) * sizeof(float), stream);
    (void)hipMemsetAsync(tstats, 0, (NC * 2) * sizeof(float), stream);

    dim3 blk(256);

    k_region_gemm<<<dim3(NROWS / 128, C / 64, 4), blk, 0, stream>>>(
        x, region_W, region_b, hbuf, rstats);

    k_region_finalize<<<(4 * BT_TOT * C) / 256, blk, 0, stream>>>(
        hbuf, rstats, region_g, region_be, hmean);

    k_heads<<<BT_TOT, 64, 0, stream>>>(hmean, upfc_W, upfc_b, midfc_W, midfc_b,
                                       d1fc_W, d1fc_b, d2fc_W, d2fc_b, out);

    k_class_pass1<<<dim3(NROWS / 128, C / 64, NC), blk, 0, stream>>>(
        hbuf, class_W, class_b, cstats);

    k_class_pass2<<<dim3(NROWS / 128, C / 64, NC), blk, 0, stream>>>(
        hbuf, class_W, class_b, cstats, class_g, class_be, f_v);

    k_hg_gemm<<<dim3((BT_TOT * NC) / 128, C / 64, 1), blk, 0, stream>>>(f_v, gat_W, hg);

    k_gat<<<BT_TOT, blk, 0, stream>>>(f_v, hg, gat_al, gat_ar, adj_mask, gbuf);

    k_tconv<<<(BT_TOT * NC * C) / 256, blk, 0, stream>>>(gbuf, tconv_W, tconv_b, vbuf, tstats);

    k_final<<<(BT_TOT * NC * C) / 256, blk, 0, stream>>>(vbuf, tstats, tbn_g, tbn_be,
                                                         out + 43008);
}